// RNASampler_67834713473405
// MI455X (gfx1250) — compile-verified
//
#include <hip/hip_runtime.h>

typedef float v2f __attribute__((ext_vector_type(2)));
typedef float v8f __attribute__((ext_vector_type(8)));

#define L_RES   96
#define NSTEPS  200
#define NPAIR   4560        // 96*95/2
#define MIN_D   3.0f
#define CONTACT 8.0f
#define EPS_F   1e-8f

// D = A(MxK) * B(KxN) + bias, optional relu. M = 96 fixed.
// Uses V_WMMA_F32_16X16X4_F32 (exact f32 math -> matches reference thresholds).
// A-frag (16x4 f32, 2 VGPR): lane<16 holds {K0,K1}, lane>=16 holds {K2,K3}, row = M=lane%16.
// B-frag (4x16 f32, 2 VGPR): mirrored (VGPR v: lanes 0-15 -> K=v, 16-31 -> K=v+2).
// C/D (16x16 f32, 8 VGPR): VGPR v: lanes 0-15 -> M=v, lanes 16-31 -> M=v+8, N=lane%16.
__device__ __forceinline__ void mlp_layer_wmma(
    const float* __restrict__ A, const float* __restrict__ B,
    const float* __restrict__ bias, float* __restrict__ Cout,
    int K, int N, bool relu, int wave, int nwaves, int lane)
{
  const int MT   = L_RES / 16;          // 6
  const int NT   = (N + 15) >> 4;
  const int half = lane >> 4;
  const int lm   = lane & 15;
  for (int t = wave; t < MT * NT; t += nwaves) {
    const int mt   = t / NT;
    const int nt   = t - mt * NT;
    const int arow = mt * 16 + lm;
    const int bcol = nt * 16 + lm;
    const bool bok = (bcol < N);        // tail mask for N=9 layer (select, not branch)
    v8f acc = {0.f, 0.f, 0.f, 0.f, 0.f, 0.f, 0.f, 0.f};
    for (int k = 0; k < K; k += 4) {
      const int k0 = k + half * 2;
      v2f a, b;
      a.x = A[arow * K + k0];
      a.y = A[arow * K + k0 + 1];
      b.x = bok ? B[(k0    ) * N + bcol] : 0.f;
      b.y = bok ? B[(k0 + 1) * N + bcol] : 0.f;
      acc = __builtin_amdgcn_wmma_f32_16x16x4_f32(
          /*neg_a=*/false, a, /*neg_b=*/false, b,
          /*c_mod=*/(short)0, acc, /*reuse_a=*/false, /*reuse_b=*/false);
    }
    if (bok) {
      const float bb = bias[bcol];
      #pragma unroll
      for (int v = 0; v < 8; ++v) {
        const int row = mt * 16 + v + half * 8;
        float val = acc[v] + bb;
        if (relu) val = fmaxf(val, 0.f);
        Cout[row * N + bcol] = val;
      }
    }
  }
}

__global__ __launch_bounds__(256)
void rna_sampler_kernel(
    const float* __restrict__ emb,   const float* __restrict__ noise,
    const float* __restrict__ quats, const float* __restrict__ trans,
    const float* __restrict__ W1, const float* __restrict__ b1,
    const float* __restrict__ W2, const float* __restrict__ b2,
    const float* __restrict__ W3, const float* __restrict__ b3,
    const float* __restrict__ motif,
    const int* __restrict__ start_pos, const int* __restrict__ do_motif,
    const int* __restrict__ motif_idx, const int* __restrict__ do_rotate,
    const int* __restrict__ do_translate,
    float* __restrict__ out, float* __restrict__ ws)
{
  __shared__ float sc [L_RES * 9];   // coords [r][a][c]
  __shared__ float scn[L_RES * 9];   // ping-pong buffer
  __shared__ float a0 [L_RES * 3];   // representative atom positions
  __shared__ unsigned char cp[NPAIR];
  __shared__ short sII[NPAIR];
  __shared__ short sJJ[NPAIR];
  __shared__ int   scnt;

  const int tid  = threadIdx.x;
  const int lane = tid & 31;
  const int wave = tid >> 5;

  // ---- triu pair index tables (np.triu_indices order: row-major) ----
  for (int i = tid; i < L_RES - 1; i += 256) {
    int base = i * (L_RES - 1) - (i * (i - 1)) / 2;
    for (int j = i + 1; j < L_RES; ++j) {
      int p = base + (j - i - 1);
      sII[p] = (short)i;
      sJJ[p] = (short)j;
    }
  }

  // ---- MLP: emb -> h1 -> h2 -> init coords (WMMA f32) ----
  __builtin_prefetch(W1, 0, 1);
  __builtin_prefetch(W2, 0, 1);
  float* h1 = ws;                    // 96*256 f32
  float* h2 = ws + L_RES * 256;      // 96*128 f32
  mlp_layer_wmma(emb, W1, b1, h1, 512, 256, /*relu=*/true,  wave, 8, lane);
  __syncthreads();
  mlp_layer_wmma(h1,  W2, b2, h2, 256, 128, /*relu=*/true,  wave, 8, lane);
  __syncthreads();
  mlp_layer_wmma(h2,  W3, b3, sc, 128,   9, /*relu=*/false, wave, 8, lane);
  __syncthreads();
  for (int idx = tid; idx < L_RES * 9; idx += 256)
    sc[idx] += 0.1f * noise[idx];    // coords = init + noise*(0.1*TEMP)
  __syncthreads();

  // ---- 200 sequential diffusion steps ----
  for (int s = 0; s < NSTEPS; ++s) {
    const int sp = start_pos[s], dm = do_motif[s], mi = motif_idx[s];
    const int dr = do_rotate[s], dt = do_translate[s];
    const float q0 = quats[s * 4 + 0], q1 = quats[s * 4 + 1];
    const float q2 = quats[s * 4 + 2], q3 = quats[s * 4 + 3];
    const float tx = trans[s * 3 + 0], ty = trans[s * 3 + 1], tz = trans[s * 3 + 2];

    const float qn = sqrtf(q0 * q0 + q1 * q1 + q2 * q2 + q3 * q3) + EPS_F;
    const float w = q0 / qn, x = q1 / qn, y = q2 / qn, z = q3 / qn;
    const float R00 = 1.f - 2.f * (y * y + z * z), R01 = 2.f * (x * y - w * z), R02 = 2.f * (x * z + w * y);
    const float R10 = 2.f * (x * y + w * z), R11 = 1.f - 2.f * (x * x + z * z), R12 = 2.f * (y * z - w * x);
    const float R20 = 2.f * (x * z - w * y), R21 = 2.f * (y * z + w * x), R22 = 1.f - 2.f * (x * x + y * y);

    // motif replace + rotate + translate, per (residue, atom): independent items
    for (int e = tid; e < L_RES * 3; e += 256) {
      const int r = e / 3, aa = e - r * 3;
      float cx = sc[e * 3 + 0], cy = sc[e * 3 + 1], cz = sc[e * 3 + 2];
      if (dm < 3 && r >= sp && r < sp + 5) {
        cx = motif[mi * 9 + aa * 3 + 0];
        cy = motif[mi * 9 + aa * 3 + 1];
        cz = motif[mi * 9 + aa * 3 + 2];
      }
      if (dr < 1) {
        const float nx = R00 * cx + R01 * cy + R02 * cz;
        const float ny = R10 * cx + R11 * cy + R12 * cz;
        const float nz = R20 * cx + R21 * cy + R22 * cz;
        cx = nx; cy = ny; cz = nz;
      }
      if (dt < 1) { cx += 0.5f * tx; cy += 0.5f * ty; cz += 0.5f * tz; }
      scn[e * 3 + 0] = cx; scn[e * 3 + 1] = cy; scn[e * 3 + 2] = cz;
    }
    __syncthreads();

    // bond constraints: vectorized, all updates read pre-update values (scn -> sc)
    for (int e = tid; e < L_RES * 3; e += 256) {
      const int r = e / 3;
      float cx = scn[e * 3 + 0], cy = scn[e * 3 + 1], cz = scn[e * 3 + 2];
      if (r < L_RES - 1) {
        const float dx = scn[e * 3 + 9]  - cx;   // next residue, same atom
        const float dy = scn[e * 3 + 10] - cy;
        const float dz = scn[e * 3 + 11] - cz;
        const float dist = sqrtf(dx * dx + dy * dy + dz * dz + 1e-12f);
        if (dist < MIN_D) {
          const float f = (MIN_D - dist) / (dist + EPS_F);
          cx += dx * f; cy += dy * f; cz += dz * f;
        }
      }
      sc[e * 3 + 0] = cx; sc[e * 3 + 1] = cy; sc[e * 3 + 2] = cz;
    }
    __syncthreads();

    // representative atoms + contact mask (parallel, uses pre-scan positions)
    for (int r = tid; r < L_RES; r += 256) {
      a0[r * 3 + 0] = sc[r * 9 + 0];
      a0[r * 3 + 1] = sc[r * 9 + 1];
      a0[r * 3 + 2] = sc[r * 9 + 2];
    }
    __syncthreads();
    for (int p = tid; p < NPAIR; p += 256) {
      const int i = sII[p], j = sJJ[p];
      const float dx = a0[j * 3 + 0] - a0[i * 3 + 0];
      const float dy = a0[j * 3 + 1] - a0[i * 3 + 1];
      const float dz = a0[j * 3 + 2] - a0[i * 3 + 2];
      const float d = sqrtf(dx * dx + dy * dy + dz * dz + 1e-12f);
      cp[p] = (d < CONTACT) ? 1 : 0;
    }
    __syncthreads();

    // strictly sequential pair scan (loop-carried dep on a0).
    // Within row i only a0[j>i] is written -> a0[i] is row-invariant, cache it.
    if (tid == 0) {
      int p = 0;
      for (int i = 0; i < L_RES - 1; ++i) {
        const float xi0 = a0[i * 3 + 0], xi1 = a0[i * 3 + 1], xi2 = a0[i * 3 + 2];
        for (int j = i + 1; j < L_RES; ++j, ++p) {
          if (!cp[p]) continue;
          const float dx = a0[j * 3 + 0] - xi0;
          const float dy = a0[j * 3 + 1] - xi1;
          const float dz = a0[j * 3 + 2] - xi2;
          const float d = sqrtf(dx * dx + dy * dy + dz * dz + 1e-12f);
          if (d < MIN_D) {
            const float f = (MIN_D - d) / (d + EPS_F);
            a0[j * 3 + 0] = xi0 + dx * f;
            a0[j * 3 + 1] = xi1 + dy * f;
            a0[j * 3 + 2] = xi2 + dz * f;
          }
        }
      }
    }
    __syncthreads();

    // write representative atoms back into coords
    for (int r = tid; r < L_RES; r += 256) {
      sc[r * 9 + 0] = a0[r * 3 + 0];
      sc[r * 9 + 1] = a0[r * 3 + 1];
      sc[r * 9 + 2] = a0[r * 3 + 2];
    }
    __syncthreads();
  }

  // ---- final contact ratio + confidence ----
  if (tid == 0) scnt = 0;
  __syncthreads();
  int local = 0;
  for (int p = tid; p < NPAIR; p += 256) {
    const int i = sII[p], j = sJJ[p];
    const float dx = sc[j * 9 + 0] - sc[i * 9 + 0];
    const float dy = sc[j * 9 + 1] - sc[i * 9 + 1];
    const float dz = sc[j * 9 + 2] - sc[i * 9 + 2];
    const float d = sqrtf(dx * dx + dy * dy + dz * dz + 1e-12f);
    local += (d < CONTACT) ? 1 : 0;
  }
  atomicAdd(&scnt, local);
  __syncthreads();

  for (int idx = tid; idx < L_RES * 9; idx += 256) out[idx] = sc[idx];
  if (tid == 0) {
    const float ratio = (2.0f * (float)scnt) / ((float)(L_RES * (L_RES - 1)) * 0.5f);
    out[L_RES * 9] = fminf(0.3f + 0.7f * ratio, 1.0f);
  }
}

extern "C" void kernel_launch(void* const* d_in, const int* in_sizes, int n_in,
                              void* d_out, int out_size, void* d_ws, size_t ws_size,
                              hipStream_t stream) {
  (void)in_sizes; (void)n_in; (void)out_size; (void)ws_size;
  // Workspace usage: h1 (96*256) + h2 (96*128) floats = 144 KiB.
  rna_sampler_kernel<<<1, 256, 0, stream>>>(
      (const float*)d_in[0],  (const float*)d_in[1],  (const float*)d_in[2],
      (const float*)d_in[3],  (const float*)d_in[4],  (const float*)d_in[5],
      (const float*)d_in[6],  (const float*)d_in[7],  (const float*)d_in[8],
      (const float*)d_in[9],  (const float*)d_in[10],
      (const int*)d_in[11], (const int*)d_in[12], (const int*)d_in[13],
      (const int*)d_in[14], (const int*)d_in[15],
      (float*)d_out, (float*)d_ws);
}